// Loss_dict_50646254354805
// MI455X (gfx1250) — compile-verified
//
#include <hip/hip_runtime.h>
#include <math.h>

// ---------------------------------------------------------------------------
// CDNA5 / gfx1250, wave32.  One fused-loss pipeline:
//   rgb MSE  +  2x interlevel (merge-based O(P+X))  +  distortion (WMMA
//   16x16x4 f32 bilinear form per ray)  +  2x hash decay (segment sums).
// All partial results atomically accumulate into d_out[0] (float32 scalar).
// d_ws usage: 2*65536 floats = 512 KB (segment sums + counts), reused for
// both hash levels via stream ordering.
// ---------------------------------------------------------------------------

typedef __attribute__((ext_vector_type(2))) float v2f;
typedef __attribute__((ext_vector_type(8))) float v8f;

#define R_ROWS   4096
#define NS       48          // samples per ray (render)
#define NB       98          // 2*NS + 2 blurred bound count
#define NUM_SEG  65536
#define M_ENC    (R_ROWS * NS)   // 196608
#define W_RGB_C  1.0f
#define W_INTER_C 1.0f
#define W_DIST_C 0.01f
#define W_HASH_C 0.1f

__device__ __forceinline__ float waveReduce32(float v) {
  #pragma unroll
  for (int o = 16; o > 0; o >>= 1) v += __shfl_xor(v, o, 32);
  return v;  // all 32 lanes hold the sum
}

// ---------------------------------------------------------------------------
__global__ void init_out_kernel(float* out, int n) {
  int i = blockIdx.x * blockDim.x + threadIdx.x;
  if (i < n) out[i] = 0.0f;
}

__global__ void zero_ws_kernel(float* ws, int n) {
  int i = blockIdx.x * blockDim.x + threadIdx.x;
  if (i < n) ws[i] = 0.0f;
}

// ---------------------------------------------------------------------------
__global__ void rgb_loss_kernel(const float* __restrict__ pd,
                                const float* __restrict__ gt,
                                float* out) {
  int i = blockIdx.x * blockDim.x + threadIdx.x;
  float d = 0.0f;
  if (i < R_ROWS * 3) { float t = pd[i] - gt[i]; d = t * t; }
  d = waveReduce32(d);
  __shared__ float s[8];
  int wid = threadIdx.x >> 5;
  if ((threadIdx.x & 31) == 0) s[wid] = d;
  __syncthreads();
  if (threadIdx.x == 0) {
    float t = 0.0f;
    for (int k = 0; k < (int)(blockDim.x >> 5); ++k) t += s[k];
    atomicAdd(out, W_RGB_C * t / (float)(R_ROWS * 3));
  }
}

// ---------------------------------------------------------------------------
// Interlevel loss: one wave per ray.  Lane 0 does the O(2N) merge of the two
// sorted shifted copies (replacing jnp.argsort) and the serial cumsums into
// per-wave LDS; then all 32 lanes binary-search the 98-entry blurred CDF for
// each proposal sample (replacing the [R,98,X] mask tensor).
// Per-wave LDS region layout (floats):
//   sd[49] | wn[48] | bounds[98] | blurw[98] | cdf[98] | xbuf[257]
#define IL_SD   0
#define IL_WN   49
#define IL_BND  (49 + 48)
#define IL_BLW  (IL_BND + NB)
#define IL_CDF  (IL_BLW + NB)
#define IL_XB   (IL_CDF + NB)
#define IL_WAVE (IL_XB + 257)     // 648 floats per wave

__global__ void inter_loss_kernel(const float* __restrict__ rsd,
                                  const float* __restrict__ rw,
                                  const float* __restrict__ psd,
                                  const float* __restrict__ pwt,
                                  int X, float pulse, float* out) {
  __shared__ float sm[8 * IL_WAVE];
  int wid  = threadIdx.x >> 5;
  int lane = threadIdx.x & 31;
  int row  = blockIdx.x * 8 + wid;

  float* sd     = sm + wid * IL_WAVE + IL_SD;
  float* wn     = sm + wid * IL_WAVE + IL_WN;
  float* bounds = sm + wid * IL_WAVE + IL_BND;
  float* blurw  = sm + wid * IL_WAVE + IL_BLW;
  float* cdf    = sm + wid * IL_WAVE + IL_CDF;
  float* xbuf   = sm + wid * IL_WAVE + IL_XB;

  const float* srow = rsd + row * (NS + 1);
  const float* wrow = rw  + row * NS;
  for (int n = lane; n <= NS; n += 32) sd[n] = srow[n];
  __syncthreads();
  for (int n = lane; n < NS; n += 32)
    wn[n] = wrow[n] / (sd[n + 1] - sd[n] + 1e-8f);
  __syncthreads();

  if (lane == 0) {
    const float inv2p = 1.0f / (2.0f * pulse);
    // stable 2-way merge of (sd - pulse) and (sd + pulse); carry radio terms.
    int i = 0, j = 0;
    for (int k = 0; k < NB; ++k) {
      float lv = (i <= NS) ? sd[i] - pulse : 1e30f;
      float rv = (j <= NS) ? sd[j] + pulse : 1e30f;
      bool takeL = (i <= NS) && (j > NS || lv <= rv);
      float val, rd;
      if (takeL) {
        float diff = (i == 0) ? wn[0] : ((i == NS) ? -wn[NS - 1] : wn[i] - wn[i - 1]);
        val = lv; rd = diff * inv2p; ++i;
      } else {
        float diff = (j == 0) ? wn[0] : ((j == NS) ? -wn[NS - 1] : wn[j] - wn[j - 1]);
        val = rv; rd = -diff * inv2p; ++j;
      }
      bounds[k] = val;
      if (k < NB - 1) xbuf[k] = rd;   // stash radio_sorted in xbuf temporarily
    }
    // recover = max(cumsum(ds * cumsum(radio)), 0); blur_w = [0, recover]
    float run1 = 0.0f, run2 = 0.0f;
    blurw[0] = 0.0f;
    for (int k = 0; k < NB - 1; ++k) {
      run1 += xbuf[k];
      run2 += (bounds[k + 1] - bounds[k]) * run1;
      blurw[k + 1] = fmaxf(run2, 0.0f);
    }
    // trapezoid CDF
    cdf[0] = 0.0f;
    for (int k = 0; k < NB - 1; ++k)
      cdf[k + 1] = cdf[k] + 0.5f * (blurw[k + 1] + blurw[k]) * (bounds[k + 1] - bounds[k]);
  }
  __syncthreads();

  // sorted_interp_quad via binary search (x sorted, xp sorted)
  const float* xr = psd + row * X;
  for (int t = lane; t < X; t += 32) {
    float x = xr[t];
    int lo = 0, hi = NB;
    while (lo < hi) { int m = (lo + hi) >> 1; if (bounds[m] <= x) lo = m + 1; else hi = m; }
    int i1 = (lo < NB) ? lo : NB - 1;
    int i0 = (lo > 0) ? lo - 1 : 0;
    float xp0 = bounds[i0], xp1 = bounds[i1];
    float f0 = blurw[i0],  f1 = blurw[i1];
    float den = xp1 - xp0;
    float off = (den != 0.0f) ? (x - xp0) / den : 0.0f;
    off = fminf(fmaxf(off, 0.0f), 1.0f);
    xbuf[t] = cdf[i0] + (x - xp0) * (f0 + f1 * off + f0 * (1.0f - off)) * 0.5f;
  }
  __syncthreads();

  float acc = 0.0f;
  const float* pr = pwt + row * (X - 1);
  for (int t = lane; t < X - 1; t += 32) {
    float ws = xbuf[t + 1] - xbuf[t];
    float p  = pr[t];
    float d  = fmaxf(ws - p, 0.0f);
    acc += d * d / (p + 1e-5f);
  }
  acc = waveReduce32(acc);
  if (lane == 0)
    atomicAdd(out, W_INTER_C * acc / (float)(R_ROWS * (X - 1)));
}

// ---------------------------------------------------------------------------
// Distortion loss: one wave per ray.  part1 = w^T K w with K[n][m]=|mid_n-mid_m|
// computed through V_WMMA_F32_16X16X4_F32: K tiled 3x3 of 16x16, K-dim chained
// in chunks of 4, B columns replicate w -> accumulator columns all hold (K w);
// lane-local dot with w + wave reduce + /16 gives the bilinear form.
__global__ void dist_loss_kernel(const float* __restrict__ rsd,
                                 const float* __restrict__ rw,
                                 float* out) {
  __shared__ float sm[8 * 144];          // per wave: mid[48] | w[48] | del[48]
  int wid  = threadIdx.x >> 5;
  int lane = threadIdx.x & 31;
  int row  = blockIdx.x * 8 + wid;
  float* mid = sm + wid * 144;
  float* w   = mid + 48;
  float* del = w + 48;

  const float* srow = rsd + row * (NS + 1);
  const float* wrow = rw  + row * NS;
  for (int n = lane; n < NS; n += 32) {
    float a = srow[n], b = srow[n + 1];
    mid[n] = 0.5f * (a + b);
    w[n]   = wrow[n];
    del[n] = b - a;
  }
  __syncthreads();

  // A layout (16x4 f32): lane L -> M = L&15; VGPR v -> K = v + 2*(L>>4).
  // B layout (4x16 f32): VGPR v -> K = v + 2*(L>>4) (column = lane&15).
  const int m_a  = lane & 15;
  const int koff = (lane >> 4) * 2;
  float part1 = 0.0f;
  #pragma unroll
  for (int ti = 0; ti < 3; ++ti) {
    v8f c = {0.f, 0.f, 0.f, 0.f, 0.f, 0.f, 0.f, 0.f};
    float ma = mid[16 * ti + m_a];
    #pragma unroll
    for (int tj = 0; tj < 3; ++tj) {
      #pragma unroll
      for (int kb = 0; kb < 16; kb += 4) {
        int k0 = 16 * tj + kb + koff;
        v2f a, b;
        a.x = fabsf(ma - mid[k0]);
        a.y = fabsf(ma - mid[k0 + 1]);
        b.x = w[k0];
        b.y = w[k0 + 1];
        c = __builtin_amdgcn_wmma_f32_16x16x4_f32(
                /*neg_a=*/false, a, /*neg_b=*/false, b,
                /*c_mod=*/(short)0, c, /*reuse_a=*/false, /*reuse_b=*/false);
      }
    }
    // C/D layout: VGPR p: lanes 0-15 -> M=p, lanes 16-31 -> M=p+8.
    float s = 0.0f;
    int mb = 16 * ti + 8 * (lane >> 4);
    #pragma unroll
    for (int p = 0; p < 8; ++p) s += c[p] * w[mb + p];
    part1 += waveReduce32(s) * (1.0f / 16.0f);  // 16 identical columns
  }

  // part2 = sum(w^2 * ds) / 3
  float s2 = w[lane] * w[lane] * del[lane];
  if (lane < 16) { int n2 = lane + 32; s2 += w[n2] * w[n2] * del[n2]; }
  s2 = waveReduce32(s2);

  if (lane == 0)
    atomicAdd(out, W_DIST_C * (fabsf(part1) + fabsf(s2 * (1.0f / 3.0f)))
                           / (float)R_ROWS);
}

// ---------------------------------------------------------------------------
__global__ void hash_accum_kernel(const float* __restrict__ e,
                                  const int* __restrict__ idx,
                                  float* sums, float* cnt, int M) {
  int i = blockIdx.x * blockDim.x + threadIdx.x;
  if (i < M) {
    float a = e[2 * i], b = e[2 * i + 1];
    int s = idx[i];
    atomicAdd(&sums[s], a * a + b * b);
    atomicAdd(&cnt[s], 1.0f);
  }
}

__global__ void hash_finalize_kernel(const float* __restrict__ sums,
                                     const float* __restrict__ cnt,
                                     float* out) {
  int i = blockIdx.x * blockDim.x + threadIdx.x;
  float v = 0.0f;
  if (i < NUM_SEG) v = sums[i] / fmaxf(cnt[i], 1.0f);
  v = waveReduce32(v);
  __shared__ float s[8];
  int wid = threadIdx.x >> 5;
  if ((threadIdx.x & 31) == 0) s[wid] = v;
  __syncthreads();
  if (threadIdx.x == 0) {
    float t = 0.0f;
    for (int k = 0; k < (int)(blockDim.x >> 5); ++k) t += s[k];
    atomicAdd(out, W_HASH_C * t / (float)(NUM_SEG * 2));
  }
}

// ---------------------------------------------------------------------------
extern "C" void kernel_launch(void* const* d_in, const int* in_sizes, int n_in,
                              void* d_out, int out_size, void* d_ws, size_t ws_size,
                              hipStream_t stream) {
  (void)in_sizes; (void)n_in; (void)ws_size;
  const float* pd   = (const float*)d_in[0];
  const float* gt   = (const float*)d_in[1];
  const float* rsd  = (const float*)d_in[2];
  const float* rw   = (const float*)d_in[3];
  const float* psd0 = (const float*)d_in[4];
  const float* pw0  = (const float*)d_in[5];
  const float* psd1 = (const float*)d_in[6];
  const float* pw1  = (const float*)d_in[7];
  const float* e0   = (const float*)d_in[8];
  const float* e1   = (const float*)d_in[9];
  const int*   id0  = (const int*)d_in[10];
  const int*   id1  = (const int*)d_in[11];
  float* out = (float*)d_out;

  float* sums = (float*)d_ws;          // NUM_SEG floats
  float* cnt  = sums + NUM_SEG;        // NUM_SEG floats (512 KB total)

  init_out_kernel<<<1, 64, 0, stream>>>(out, out_size);

  rgb_loss_kernel<<<(R_ROWS * 3 + 255) / 256, 256, 0, stream>>>(pd, gt, out);

  inter_loss_kernel<<<R_ROWS / 8, 256, 0, stream>>>(rsd, rw, psd0, pw0, 257, 0.01f,  out);
  inter_loss_kernel<<<R_ROWS / 8, 256, 0, stream>>>(rsd, rw, psd1, pw1, 97,  0.005f, out);

  dist_loss_kernel<<<R_ROWS / 8, 256, 0, stream>>>(rsd, rw, out);

  // hash level 0 then level 1, reusing scratch (stream-ordered)
  zero_ws_kernel<<<(2 * NUM_SEG + 255) / 256, 256, 0, stream>>>(sums, 2 * NUM_SEG);
  hash_accum_kernel<<<(M_ENC + 255) / 256, 256, 0, stream>>>(e0, id0, sums, cnt, M_ENC);
  hash_finalize_kernel<<<NUM_SEG / 256, 256, 0, stream>>>(sums, cnt, out);

  zero_ws_kernel<<<(2 * NUM_SEG + 255) / 256, 256, 0, stream>>>(sums, 2 * NUM_SEG);
  hash_accum_kernel<<<(M_ENC + 255) / 256, 256, 0, stream>>>(e1, id1, sums, cnt, M_ENC);
  hash_finalize_kernel<<<NUM_SEG / 256, 256, 0, stream>>>(sums, cnt, out);
}